// nnChamferLoss_33930241639080
// MI455X (gfx1250) — compile-verified
//
#include <hip/hip_runtime.h>

typedef float v2f __attribute__((ext_vector_type(2)));
typedef float v8f __attribute__((ext_vector_type(8)));

#define FLT_BIG 3.402823466e+38f

// One wave (32 threads) per 16-row tile of A. Sweeps all of B with
// V_WMMA_F32_16X16X4_F32 (K=8 as two k-steps), keeping a running per-column
// min in the 8 accumulator VGPRs, then reduces rows cross-lane at the end.
// gridDim.y selects the pass: pass0 A=p0,B=p1 (dist0), pass1 swapped (dist1).
__global__ __launch_bounds__(32)
void chamfer_pass_kernel(const float* __restrict__ p0,
                         const float* __restrict__ p1,
                         int n0, int n1,
                         const int* __restrict__ truncPtr,
                         float* __restrict__ partials,
                         int maxTiles)
{
    const int pass = blockIdx.y;
    const float* __restrict__ A = (pass == 0) ? p0 : p1;
    const float* __restrict__ B = (pass == 0) ? p1 : p0;
    const int NA = (pass == 0) ? n0 : n1;
    const int NB = (pass == 0) ? n1 : n0;

    const int tile  = blockIdx.x;
    const int ibase = tile << 4;
    const int lane  = threadIdx.x;   // 0..31 (wave32)
    const int col   = lane & 15;
    const bool hi   = lane >= 16;

    const int pidx = 2 * (pass * maxTiles + tile);

    if (ibase >= NA) {               // uniform across the wave
        if (lane == 0) { partials[pidx] = 0.0f; partials[pidx + 1] = 0.0f; }
        return;
    }

    // ---- Hoisted A' fragment: A'[i] = (-2ax, -2ay, -2az, |a|^2, 1, 0, 0, 0)
    // 32-bit 16x4 A layout: lanes 0-15 hold (K0,K1); lanes 16-31 hold (K2,K3).
    int ia = ibase + col; if (ia > NA - 1) ia = NA - 1;
    const float ax = A[3 * ia + 0], ay = A[3 * ia + 1], az = A[3 * ia + 2];
    const float a2 = ax * ax + ay * ay + az * az;
    v2f afrag0, afrag1;
    afrag0.x = hi ? (-2.0f * az) : (-2.0f * ax);
    afrag0.y = hi ? a2           : (-2.0f * ay);
    afrag1.x = hi ? 0.0f : 1.0f;   // K4 = 1 (low half), K5..7 = 0
    afrag1.y = 0.0f;

    float rmin[8];
#pragma unroll
    for (int k = 0; k < 8; ++k) rmin[k] = FLT_BIG;

    for (int jb = 0; jb < NB; jb += 16) {
        int j = jb + col; if (j > NB - 1) j = NB - 1;  // dup cols are min-safe
        const float bx = B[3 * j + 0], by = B[3 * j + 1], bz = B[3 * j + 2];
        const float b2 = bx * bx + by * by + bz * bz;
        // B'[j] = (bx, by, bz, 1, |b|^2, 0, 0, 0); mirrored 4x16 layout:
        // lanes 0-15 hold (K0,K1), lanes 16-31 hold (K2,K3) for column lane%16.
        v2f bfrag0, bfrag1;
        bfrag0.x = hi ? bz   : bx;
        bfrag0.y = hi ? 1.0f : by;
        bfrag1.x = hi ? 0.0f : b2;
        bfrag1.y = 0.0f;

        v8f c = {};
        v8f d = __builtin_amdgcn_wmma_f32_16x16x4_f32(
                    false, afrag0, false, bfrag0, (short)0, c, false, false);
        d = __builtin_amdgcn_wmma_f32_16x16x4_f32(
                    false, afrag1, false, bfrag1, (short)0, d, false, false);

#pragma unroll
        for (int k = 0; k < 8; ++k) rmin[k] = fminf(rmin[k], d[k]);
    }

    // Per-row min over columns: reduce across the 16 lanes of each half.
    // (D layout: lane holds column N=lane%16; VGPR r holds row r + 8*(lane>=16))
#pragma unroll
    for (int k = 0; k < 8; ++k) {
        float v = rmin[k];
        v = fminf(v, __shfl_xor(v, 1, 32));
        v = fminf(v, __shfl_xor(v, 2, 32));
        v = fminf(v, __shfl_xor(v, 4, 32));
        v = fminf(v, __shfl_xor(v, 8, 32));
        rmin[k] = fmaxf(v, 0.0f);   // max(d,0) commutes with min over j
    }

    const float t = (float)(*truncPtr);
    const bool useTrunc = t > 0.0f;
    float s = 0.0f, cnt = 0.0f;
#pragma unroll
    for (int k = 0; k < 8; ++k) {
        const int row = ibase + k + (hi ? 8 : 0);
        const bool ok = (row < NA) && (!useTrunc || rmin[k] <= t);
        s   += ok ? rmin[k] : 0.0f;
        cnt += ok ? 1.0f : 0.0f;
    }
    // low half holds rows 0-7, high half rows 8-15; combine across halves
    s   += __shfl_xor(s,   16, 32);
    cnt += __shfl_xor(cnt, 16, 32);

    if (lane == 0) { partials[pidx] = s; partials[pidx + 1] = cnt; }
}

// Single-block deterministic tree reduction -> mean0 + mean1
__global__ __launch_bounds__(256)
void chamfer_reduce_kernel(const float* __restrict__ partials,
                           int nT0, int nT1, int maxTiles,
                           float* __restrict__ out)
{
    __shared__ float s0s[256], c0s[256], s1s[256], c1s[256];
    const int tid = threadIdx.x;
    float s0 = 0.0f, c0 = 0.0f, s1 = 0.0f, c1 = 0.0f;
    for (int i = tid; i < nT0; i += 256) {
        s0 += partials[2 * i];
        c0 += partials[2 * i + 1];
    }
    for (int i = tid; i < nT1; i += 256) {
        s1 += partials[2 * (maxTiles + i)];
        c1 += partials[2 * (maxTiles + i) + 1];
    }
    s0s[tid] = s0; c0s[tid] = c0; s1s[tid] = s1; c1s[tid] = c1;
    __syncthreads();
    for (int off = 128; off > 0; off >>= 1) {
        if (tid < off) {
            s0s[tid] += s0s[tid + off]; c0s[tid] += c0s[tid + off];
            s1s[tid] += s1s[tid + off]; c1s[tid] += c1s[tid + off];
        }
        __syncthreads();
    }
    if (tid == 0) {
        const float m0 = s0s[0] / fmaxf(c0s[0], 1.0f);
        const float m1 = s1s[0] / fmaxf(c1s[0], 1.0f);
        out[0] = m0 + m1;
    }
}

extern "C" void kernel_launch(void* const* d_in, const int* in_sizes, int n_in,
                              void* d_out, int out_size, void* d_ws, size_t ws_size,
                              hipStream_t stream) {
    const float* p0 = (const float*)d_in[0];
    const float* p1 = (const float*)d_in[1];
    const int* truncPtr = (const int*)d_in[2];

    const int n0 = in_sizes[0] / 3;   // 16384
    const int n1 = in_sizes[1] / 3;   // 16384
    const int nT0 = (n0 + 15) / 16;
    const int nT1 = (n1 + 15) / 16;
    const int maxTiles = (nT0 > nT1) ? nT0 : nT1;

    float* partials = (float*)d_ws;   // 2 * maxTiles * 2 floats (16 KB)

    dim3 grid(maxTiles, 2, 1);
    chamfer_pass_kernel<<<grid, dim3(32, 1, 1), 0, stream>>>(
        p0, p1, n0, n1, truncPtr, partials, maxTiles);

    chamfer_reduce_kernel<<<1, 256, 0, stream>>>(
        partials, nT0, nT1, maxTiles, (float*)d_out);
}